// ETGNN_8366596292958
// MI455X (gfx1250) — compile-verified
//
#include <hip/hip_runtime.h>
#include <math.h>

// ---------------------------------------------------------------------------
// ETGNN forward for MI455X (gfx1250, wave32).
// GEMMs use V_WMMA_F32_16X16X4_F32 (full fp32 precision; compute is nowhere
// near the bottleneck -> no reason to downcast). Edge softmax/scatter uses
// L2-resident float atomics (agg is 25.6MB << 192MB L2).
// ---------------------------------------------------------------------------

typedef __attribute__((ext_vector_type(2))) float v2f;
typedef __attribute__((ext_vector_type(8))) float v8f;

#define CVi 3
#define CN 50000
#define CE 800000
#define CIN 256
#define CHID 128
#define COUT 128
#define CEH 256
#define CEO 64

// ---------------- generic fp32 WMMA GEMM:  C = act(A[M,K] @ B[K,Nc] + bias) --
// One wave per 16x16 output tile. M % 16 == 0, K % 4 == 0, Nc % 16 == 0.
// act: 0 = none, 1 = relu
__global__ __launch_bounds__(256)
void etgnn_gemm_wmma_f32(const float* __restrict__ A, const float* __restrict__ B,
                         const float* __restrict__ bias, float* __restrict__ C,
                         int M, int K, int Nc, int act) {
  const int lane = threadIdx.x & 31;
  const int wid  = threadIdx.x >> 5;
  const int tilesN = Nc >> 4;
  const int tiles  = (M >> 4) * tilesN;
  const int tile = blockIdx.x * (blockDim.x >> 5) + wid;
  if (tile >= tiles) return;                 // wave-uniform: EXEC stays all-1s for WMMA

  const int tm   = tile / tilesN;
  const int tn   = tile - tm * tilesN;
  const int half = lane >> 4;                // 0: lanes 0-15, 1: lanes 16-31
  const int l16  = lane & 15;
  const int row  = (tm << 4) + l16;          // A-matrix row for this lane (M = lane&15)
  const int col  = (tn << 4) + l16;          // B/C column for this lane (N = lane&15)
  const int koff = half << 1;                // K sub-offset: {0,1} or {2,3}

  const float* Arow = A + (size_t)row * K + koff;           // b64 loads (8B aligned)
  const float* Bcol = B + (size_t)koff * Nc + col;

  v8f acc = {};
#pragma unroll 4
  for (int k = 0; k < K; k += 4) {
    v2f a = *(const v2f*)(Arow + k);                        // A[row][k+koff .. +1]
    v2f b;
    b.x = Bcol[(size_t)k * Nc];                             // B[k+koff  ][col]
    b.y = Bcol[(size_t)(k + 1) * Nc];                       // B[k+koff+1][col]
    acc = __builtin_amdgcn_wmma_f32_16x16x4_f32(
        /*neg_a=*/false, a, /*neg_b=*/false, b,
        /*c_mod=*/(short)0, acc, /*reuse_a=*/false, /*reuse_b=*/false);
  }

  const float bv = bias ? bias[col] : 0.0f;
  const int r0 = (tm << 4) + (half << 3);    // C/D layout: VGPR r -> M = r + 8*half
#pragma unroll
  for (int r = 0; r < 8; ++r) {
    float v = acc[r] + bv;
    if (act == 1) v = fmaxf(v, 0.0f);
    C[(size_t)(r0 + r) * Nc + col] = v;
  }
}

// ---------------- s[n] = dot(z[n, 0:128], tw[0:128]) ------------------------
__global__ __launch_bounds__(256)
void etgnn_node_score(const float* __restrict__ z, const float* __restrict__ tw,
                      float* __restrict__ s, int N) {
  const int lane = threadIdx.x & 31;
  const int node = (int)((blockIdx.x * blockDim.x + threadIdx.x) >> 5);
  if (node >= N) return;
  const float* zr = z + (size_t)node * 128;
  float v = zr[lane]      * tw[lane]
          + zr[lane + 32] * tw[lane + 32]
          + zr[lane + 64] * tw[lane + 64]
          + zr[lane + 96] * tw[lane + 96];
  for (int o = 16; o > 0; o >>= 1) v += __shfl_xor(v, o, 32);
  if (lane == 0) s[node] = v;
}

// ---------------- fill ------------------------------------------------------
__global__ void etgnn_fill_f32(float* __restrict__ p, float v, long n) {
  long i = (long)blockIdx.x * blockDim.x + threadIdx.x;
  if (i < n) p[i] = v;
}

// ---------------- edge pass A: logit = exp(s[src]*(-|dt|)/500), seg-max -----
// logits are strictly positive -> uint-ordered atomicMax with identity 0.
__global__ void etgnn_edge_logit_max(const int* __restrict__ src, const int* __restrict__ dst,
                                     const float* __restrict__ t, const float* __restrict__ s,
                                     float* __restrict__ ebuf, float* __restrict__ m, int E) {
  int e = blockIdx.x * blockDim.x + threadIdx.x;
  if (e >= E) return;
  const int sn = src[e], dn = dst[e];
  const float ew = -fabsf(t[sn] - t[dn]);
  const float l  = expf(s[sn] * ew * (1.0f / 500.0f));
  ebuf[e] = l;
  atomicMax((unsigned int*)(m + dn), __float_as_uint(l));
}

// ---------------- edge pass B: p = exp(logit - m[dst]); seg-sum -------------
__global__ void etgnn_edge_p_sum(const int* __restrict__ dst, const float* __restrict__ m,
                                 float* __restrict__ ebuf, float* __restrict__ denom, int E) {
  int e = blockIdx.x * blockDim.x + threadIdx.x;
  if (e >= E) return;
  const int dn = dst[e];
  const float p = expf(ebuf[e] - m[dn]);
  ebuf[e] = p;
  atomicAdd(denom + dn, p);
}

// ---------------- edge pass C: alpha = p / denom[dst] (denom==0 -> 1) -------
__global__ void etgnn_edge_alpha(const int* __restrict__ dst, const float* __restrict__ denom,
                                 float* __restrict__ ebuf, int E) {
  int e = blockIdx.x * blockDim.x + threadIdx.x;
  if (e >= E) return;
  float d = denom[dst[e]];
  d = (d == 0.0f) ? 1.0f : d;
  ebuf[e] = ebuf[e] / d;
}

// ---------------- scatter: agg[dst] += alpha * z[src]  (E*128 threads) ------
__global__ void etgnn_edge_scatter(const int* __restrict__ src, const int* __restrict__ dst,
                                   const float* __restrict__ alpha, const float* __restrict__ z,
                                   float* __restrict__ agg, long total) {
  long i = (long)blockIdx.x * blockDim.x + threadIdx.x;
  if (i >= total) return;
  const int e = (int)(i >> 7);
  const int c = (int)(i & 127);
  const float a = alpha[e];
  atomicAdd(agg + (size_t)dst[e] * 128 + c, a * z[(size_t)src[e] * 128 + c]);
}

// ---------------- elementwise ELU -------------------------------------------
__global__ void etgnn_elu(const float* __restrict__ in, float* __restrict__ out, long n) {
  long i = (long)blockIdx.x * blockDim.x + threadIdx.x;
  if (i >= n) return;
  const float x = in[i];
  out[i] = (x > 0.0f) ? x : expm1f(x);
}

// ---------------------------------------------------------------------------
static inline int cdiv_l(long a, long b) { return (int)((a + b - 1) / b); }

extern "C" void kernel_launch(void* const* d_in, const int* in_sizes, int n_in,
                              void* d_out, int out_size, void* d_ws, size_t ws_size,
                              hipStream_t stream) {
  (void)in_sizes; (void)n_in; (void)out_size; (void)ws_size;

  const float* features = (const float*)d_in[0];   // [V,N,IN]
  const float* t_all    = (const float*)d_in[1];   // [V,N]
  const int*   src_all  = (const int*)d_in[2];     // [V,E]
  const int*   dst_all  = (const int*)d_in[3];     // [V,E]
  const float* fc1_w    = (const float*)d_in[4];   // [IN,HID]
  const float* tfc1_w   = (const float*)d_in[5];   // [HID]
  const float* fc2_w    = (const float*)d_in[6];   // [HID,OUT]
  const float* tfc2_w   = (const float*)d_in[7];   // [OUT]
  const float* e_w1     = (const float*)d_in[8];   // [V,OUT,EH]
  const float* e_b1     = (const float*)d_in[9];   // [V,EH]
  const float* e_w2     = (const float*)d_in[10];  // [V,EH,EO]
  const float* e_b2     = (const float*)d_in[11];  // [V,EO]

  float* out_part = (float*)d_out;                         // [V,N,EO]
  float* emb_part = out_part + (size_t)CVi * CN * CEO;     // [V,N,OUT]

  // workspace layout (floats): ~132 MB total
  float* ws   = (float*)d_ws;
  float* z    = ws;  ws += (size_t)CN * CHID;   // z for both layers (HID==OUT==128)
  float* agg  = ws;  ws += (size_t)CN * CHID;   // layer-1 aggregation
  float* h1   = ws;  ws += (size_t)CN * CHID;   // elu(agg)
  float* hid  = ws;  ws += (size_t)CN * CEH;    // EDNN hidden
  float* m    = ws;  ws += CN;                  // segment max
  float* den  = ws;  ws += CN;                  // segment sum
  float* sc   = ws;  ws += CN;                  // node score
  float* ebuf = ws;  ws += CE;                  // per-edge logit -> p -> alpha

  const dim3 B(256);
  const long NH  = (long)CN * CHID;             // 6.4M
  const long EHn = (long)CE * 128;              // 102.4M scatter threads

  for (int v = 0; v < CVi; ++v) {
    const float* fv   = features + (size_t)v * CN * CIN;
    const float* tv   = t_all    + (size_t)v * CN;
    const int*   sv   = src_all  + (size_t)v * CE;
    const int*   dv   = dst_all  + (size_t)v * CE;
    float*       embv = emb_part + (size_t)v * CN * COUT;  // layer-2 agg target
    float*       outv = out_part + (size_t)v * CN * CEO;

    // ---------------- layer 1 ----------------
    etgnn_gemm_wmma_f32<<<cdiv_l((long)(CN / 16) * (CHID / 16), 8), B, 0, stream>>>(
        fv, fc1_w, nullptr, z, CN, CIN, CHID, 0);
    etgnn_node_score<<<cdiv_l(CN, 8), B, 0, stream>>>(z, tfc1_w, sc, CN);
    etgnn_fill_f32<<<cdiv_l(CN, 256), B, 0, stream>>>(m,   0.0f, CN);
    etgnn_fill_f32<<<cdiv_l(CN, 256), B, 0, stream>>>(den, 0.0f, CN);
    etgnn_fill_f32<<<cdiv_l(NH, 256), B, 0, stream>>>(agg, 0.0f, NH);
    etgnn_edge_logit_max<<<cdiv_l(CE, 256), B, 0, stream>>>(sv, dv, tv, sc, ebuf, m, CE);
    etgnn_edge_p_sum<<<cdiv_l(CE, 256), B, 0, stream>>>(dv, m, ebuf, den, CE);
    etgnn_edge_alpha<<<cdiv_l(CE, 256), B, 0, stream>>>(dv, den, ebuf, CE);
    etgnn_edge_scatter<<<cdiv_l(EHn, 256), B, 0, stream>>>(sv, dv, ebuf, z, agg, EHn);
    etgnn_elu<<<cdiv_l(NH, 256), B, 0, stream>>>(agg, h1, NH);

    // ---------------- layer 2 (agg -> emb output slice directly) ----------------
    etgnn_gemm_wmma_f32<<<cdiv_l((long)(CN / 16) * (COUT / 16), 8), B, 0, stream>>>(
        h1, fc2_w, nullptr, z, CN, CHID, COUT, 0);
    etgnn_node_score<<<cdiv_l(CN, 8), B, 0, stream>>>(z, tfc2_w, sc, CN);
    etgnn_fill_f32<<<cdiv_l(CN, 256), B, 0, stream>>>(m,   0.0f, CN);
    etgnn_fill_f32<<<cdiv_l(CN, 256), B, 0, stream>>>(den, 0.0f, CN);
    etgnn_fill_f32<<<cdiv_l(NH, 256), B, 0, stream>>>(embv, 0.0f, NH);
    etgnn_edge_logit_max<<<cdiv_l(CE, 256), B, 0, stream>>>(sv, dv, tv, sc, ebuf, m, CE);
    etgnn_edge_p_sum<<<cdiv_l(CE, 256), B, 0, stream>>>(dv, m, ebuf, den, CE);
    etgnn_edge_alpha<<<cdiv_l(CE, 256), B, 0, stream>>>(dv, den, ebuf, CE);
    etgnn_edge_scatter<<<cdiv_l(EHn, 256), B, 0, stream>>>(sv, dv, ebuf, z, embv, EHn);

    // ---------------- per-view EDNN ----------------
    etgnn_gemm_wmma_f32<<<cdiv_l((long)(CN / 16) * (CEH / 16), 8), B, 0, stream>>>(
        embv, e_w1 + (size_t)v * COUT * CEH, e_b1 + (size_t)v * CEH, hid,
        CN, COUT, CEH, 1 /*relu*/);
    etgnn_gemm_wmma_f32<<<cdiv_l((long)(CN / 16) * (CEO / 16), 8), B, 0, stream>>>(
        hid, e_w2 + (size_t)v * CEH * CEO, e_b2 + (size_t)v * CEO, outv,
        CN, CEH, CEO, 0);
  }
}